// ConditionalSigKerMMDDiscriminator_28664611733746
// MI455X (gfx1250) — compile-verified
//
#include <hip/hip_runtime.h>
#include <math.h>

typedef __attribute__((ext_vector_type(2))) float v2f;
typedef __attribute__((ext_vector_type(8))) float v8f;

#define SIGMA1_INV 1.0f   // 1/sigma
#define LAMBD      1.0f

// ---------------------------------------------------------------------------
// Kernel 1: per-(n,m)-pair fused static-Gram (WMMA fp32 16x16x4) + Goursat PDE
// grid = (4096, 4): blockIdx.x = pair (n*64+m), blockIdx.y = gram id
// block = 32 threads (one wave32) -> barriers are S_NOPs, EXEC all-ones at WMMA
// ---------------------------------------------------------------------------
__launch_bounds__(32)
__global__ void sig_pair_kernel(const float* __restrict__ x,
                                const float* __restrict__ y,
                                const float* __restrict__ z,
                                float* __restrict__ ws)
{
    __shared__ float Xs[48 * 9];      // 48 rows (zero-padded past 33) x 8 dims, stride 9
    __shared__ float Ys[48 * 9];
    __shared__ float x2s[48];
    __shared__ float y2s[48];
    __shared__ float G[48 * 49];      // padded static-kernel gram, stride 49
    __shared__ float inc[32 * 33];    // quarter double-increments, stride 33

    const int lane = threadIdx.x;
    const int pair = blockIdx.x;
    const int n = pair >> 6;
    const int m = pair & 63;
    const int g = blockIdx.y;

    const float* Aseq; const float* Bseq; float* dst; bool rbf;
    if (g == 0)      { Aseq = x; Bseq = x; dst = ws;         rbf = true;  }
    else if (g == 1) { Aseq = y; Bseq = y; dst = ws + 4096;  rbf = false; }
    else if (g == 2) { Aseq = z; Bseq = z; dst = ws + 8192;  rbf = false; }
    else             { Aseq = y; Bseq = z; dst = ws + 12288; rbf = false; }

    const float* Xg = Aseq + n * 33 * 8;
    const float* Yg = Bseq + m * 33 * 8;

    // Stage sequences into LDS, zero-padding rows 33..47 so WMMA padding is benign.
    for (int idx = lane; idx < 48 * 8; idx += 32) {
        int r = idx >> 3, c = idx & 7;
        float xv = (r < 33) ? Xg[r * 8 + c] : 0.0f;
        float yv = (r < 33) ? Yg[r * 8 + c] : 0.0f;
        Xs[r * 9 + c] = xv;
        Ys[r * 9 + c] = yv;
    }
    __syncthreads();

    // Squared norms (cheap; computed unconditionally to keep epilogue branch-light).
    for (int s = lane; s < 48; s += 32) {
        float sx = 0.f, sy = 0.f;
        #pragma unroll
        for (int d = 0; d < 8; ++d) {
            float a = Xs[s * 9 + d]; sx += a * a;
            float b = Ys[s * 9 + d]; sy += b * b;
        }
        x2s[s] = sx; y2s[s] = sy;
    }
    __syncthreads();

    // --- Static gram via V_WMMA_F32_16X16X4_F32 (K=8 split into two K=4 ops) ---
    // A 16x4 f32 layout: lanes 0-15 -> M rows, v0=K(2*half), v1=K(2*half+1)
    // B 4x16 f32 layout: lanes hold N cols, same K grouping. B[k][N] = Y[N][k].
    const int half = lane >> 4;
    const int lm16 = lane & 15;
    for (int tr = 0; tr < 3; ++tr) {
        int Ma = tr * 16 + lm16;
        v2f a1, a2;
        a1[0] = Xs[Ma * 9 + 2 * half + 0];
        a1[1] = Xs[Ma * 9 + 2 * half + 1];
        a2[0] = Xs[Ma * 9 + 4 + 2 * half + 0];
        a2[1] = Xs[Ma * 9 + 4 + 2 * half + 1];
        for (int tc = 0; tc < 3; ++tc) {
            int Nb = tc * 16 + lm16;
            v2f b1, b2;
            b1[0] = Ys[Nb * 9 + 2 * half + 0];
            b1[1] = Ys[Nb * 9 + 2 * half + 1];
            b2[0] = Ys[Nb * 9 + 4 + 2 * half + 0];
            b2[1] = Ys[Nb * 9 + 4 + 2 * half + 1];
            v8f c = {0.f, 0.f, 0.f, 0.f, 0.f, 0.f, 0.f, 0.f};
            c = __builtin_amdgcn_wmma_f32_16x16x4_f32(false, a1, false, b1,
                                                      (short)0, c, false, false);
            c = __builtin_amdgcn_wmma_f32_16x16x4_f32(false, a2, false, b2,
                                                      (short)0, c, false, false);
            // C/D layout: VGPR r -> M = r + 8*half (per half-wave), N = lane%16.
            // G is padded (48x49) -> unconditional stores, no per-element exec masking.
            const int Ng = tc * 16 + lm16;
            const int Mb = tr * 16 + 8 * half;
            if (rbf) {   // uniform per-block branch
                #pragma unroll
                for (int r = 0; r < 8; ++r) {
                    int Mg = Mb + r;
                    float d2 = x2s[Mg] + y2s[Ng] - 2.0f * c[r];
                    G[Mg * 49 + Ng] = __expf(-d2 * SIGMA1_INV);
                }
            } else {
                #pragma unroll
                for (int r = 0; r < 8; ++r) {
                    int Mg = Mb + r;
                    G[Mg * 49 + Ng] = c[r];
                }
            }
        }
    }
    __syncthreads();

    // Double increments on the 32x32 base grid; dyadic f=2 => inc64 = inc32[p/2][q/2]/4
    for (int idx = lane; idx < 32 * 32; idx += 32) {
        int p = idx >> 5, q = idx & 31;
        float v = G[(p + 1) * 49 + (q + 1)] - G[(p + 1) * 49 + q]
                - G[p * 49 + (q + 1)] + G[p * 49 + q];
        inc[p * 33 + q] = 0.25f * v;
    }
    __syncthreads();

    // --- Goursat PDE wavefront on 64x64 interior, 2 adjacent columns per lane ---
    // Lane l owns columns jA=2l+1, jB=2l+2 of k (boundary row/col = 1).
    // Per diagonal d: cell (i=d-j, j) = (k[i][j-1]+k[i-1][j])*C1 - k[i-1][j-1]*C2
    // jB's left neighbor is jA (in-lane); jA's left is lane (l-1)'s jB (shfl_up).
    const int jA = 2 * lane + 1;
    const int jB = jA + 1;
    float m1A = 1.f, m2A = 1.f, m1B = 1.f, m2B = 1.f;
    for (int d = 2; d <= 128; ++d) {
        float lm1 = __shfl_up(m1B, 1, 32);   // column 2l value on diag d-1
        float lm2 = __shfl_up(m2B, 1, 32);   // ... on diag d-2
        if (lane == 0) { lm1 = 1.f; lm2 = 1.f; }   // boundary column 0
        bool actA = (d >= jA + 1) && (d <= jA + 64);
        bool actB = (d >= jB + 1) && (d <= jB + 64);
        int iaA = d - jA - 1; iaA = iaA < 0 ? 0 : (iaA > 63 ? 63 : iaA);
        int iaB = d - jB - 1; iaB = iaB < 0 ? 0 : (iaB > 63 ? 63 : iaB);
        // (j-1)>>1 == lane for both columns
        float incA = inc[(iaA >> 1) * 33 + lane];
        float incB = inc[(iaB >> 1) * 33 + lane];
        float c1A = 1.f + 0.5f * incA + incA * incA * (1.f / 12.f);
        float c2A = 1.f - incA * incA * (1.f / 12.f);
        float c1B = 1.f + 0.5f * incB + incB * incB * (1.f / 12.f);
        float c2B = 1.f - incB * incB * (1.f / 12.f);
        float newA = (lm1 + m1A) * c1A - lm2 * c2A;
        float newB = (m1A + m1B) * c1B - m2A * c2B;   // uses pre-update m1A/m2A
        if (actA) { m2A = m1A; m1A = newA; }
        if (actB) { m2B = m1B; m1B = newB; }
    }
    if (lane == 31) dst[pair] = m1B;   // k[64][64]
}

// ---------------------------------------------------------------------------
// Kernel 2: Gauss-Jordan inverse of (K + lambda*I), 64x64, one 256-thread block
// ---------------------------------------------------------------------------
__launch_bounds__(256)
__global__ void invert_kernel(float* __restrict__ ws)
{
    __shared__ float aug[64 * 129];
    __shared__ float colk[64];
    const float* K = ws;
    float* Kinv = ws + 16384;
    const int tid = threadIdx.x;

    for (int idx = tid; idx < 64 * 128; idx += 256) {
        int r = idx >> 7, c = idx & 127;
        float v;
        if (c < 64) v = K[r * 64 + c] + ((r == c) ? LAMBD : 0.f);
        else        v = ((c - 64) == r) ? 1.f : 0.f;
        aug[r * 129 + c] = v;
    }
    __syncthreads();

    for (int k = 0; k < 64; ++k) {
        float pivot = aug[k * 129 + k];
        if (tid < 64) colk[tid] = aug[tid * 129 + k];
        __syncthreads();
        float invp = 1.0f / pivot;
        for (int c = tid; c < 128; c += 256) aug[k * 129 + c] *= invp;
        __syncthreads();
        for (int idx = tid; idx < 64 * 128; idx += 256) {
            int r = idx >> 7, c = idx & 127;
            if (r != k) aug[r * 129 + c] -= colk[r] * aug[k * 129 + c];
        }
        __syncthreads();
    }
    for (int idx = tid; idx < 4096; idx += 256) {
        int r = idx >> 6, c = idx & 63;
        Kinv[idx] = aug[r * 129 + 64 + c];
    }
}

// ---------------------------------------------------------------------------
// Kernel 3: B = Kinv*K*Kinv ; out = sum_ij B[i][j]*(Lg + Lt - 2*Lm)[j][i]
// (tr(K*Kinv*L*Kinv) = tr((Kinv*K*Kinv)*L), linear in L)
// ---------------------------------------------------------------------------
__launch_bounds__(256)
__global__ void finalize_kernel(const float* __restrict__ ws, float* __restrict__ out)
{
    __shared__ float Ks[64 * 65];
    __shared__ float Ki[64 * 65];
    __shared__ float T1[64 * 65];
    __shared__ float red[256];
    const float* K  = ws;
    const float* Lg = ws + 4096;
    const float* Lt = ws + 8192;
    const float* Lm = ws + 12288;
    const float* Kv = ws + 16384;
    const int tid = threadIdx.x;

    for (int idx = tid; idx < 4096; idx += 256) {
        int r = idx >> 6, c = idx & 63;
        Ks[r * 65 + c] = K[idx];
        Ki[r * 65 + c] = Kv[idx];
    }
    __syncthreads();
    for (int idx = tid; idx < 4096; idx += 256) {
        int i = idx >> 6, j = idx & 63;
        float s = 0.f;
        for (int k2 = 0; k2 < 64; ++k2) s += Ks[i * 65 + k2] * Ki[k2 * 65 + j];
        T1[i * 65 + j] = s;
    }
    __syncthreads();
    float acc = 0.f;
    for (int idx = tid; idx < 4096; idx += 256) {
        int i = idx >> 6, j = idx & 63;
        float b = 0.f;
        for (int k2 = 0; k2 < 64; ++k2) b += Ki[i * 65 + k2] * T1[k2 * 65 + j];
        int ji = j * 64 + i;
        float ls = Lg[ji] + Lt[ji] - 2.0f * Lm[ji];
        acc += b * ls;
    }
    red[tid] = acc;
    __syncthreads();
    for (int off = 128; off > 0; off >>= 1) {
        if (tid < off) red[tid] += red[tid + off];
        __syncthreads();
    }
    if (tid == 0) out[0] = red[0];
}

// ---------------------------------------------------------------------------
extern "C" void kernel_launch(void* const* d_in, const int* in_sizes, int n_in,
                              void* d_out, int out_size, void* d_ws, size_t ws_size,
                              hipStream_t stream)
{
    const float* x = (const float*)d_in[0];
    const float* y = (const float*)d_in[1];
    const float* z = (const float*)d_in[2];
    float* ws  = (float*)d_ws;
    float* out = (float*)d_out;

    // ws layout (floats): [0)K [4096)Lgen [8192)Ltrue [12288)Lmix [16384)Kinv
    sig_pair_kernel<<<dim3(4096, 4, 1), dim3(32, 1, 1), 0, stream>>>(x, y, z, ws);
    invert_kernel<<<dim3(1, 1, 1), dim3(256, 1, 1), 0, stream>>>(ws);
    finalize_kernel<<<dim3(1, 1, 1), dim3(256, 1, 1), 0, stream>>>(ws, out);
}